// MVSEM_15530601742424
// MI455X (gfx1250) — compile-verified
//
#include <hip/hip_runtime.h>

// Problem constants (must match reference)
#define B_    16
#define T_    512
#define V_    32000
#define D_    512
#define NM_   8
#define H_    8
#define OUT_  2048
#define DH_   64
#define LN_EPS 1e-5f

// LDS row strides padded to avoid bank conflicts on row-per-lane access:
// stride 516 -> lane l hits bank (l*4)%64 (all distinct for l in 0..15,
// +2 banks for the hi half-wave), stride 1540 likewise.
#define STRD_A 516
#define STRD_Q 1540

typedef __attribute__((ext_vector_type(2))) float v2f;
typedef __attribute__((ext_vector_type(8))) float v8f;

// ---------------------------------------------------------------------------
// WMMA helpers (CDNA5 V_WMMA_F32_16X16X4_F32, wave32)
//   A: 16x4  -> 2 VGPRs/lane: lane = M + 16*(K/2), vgpr = K%2
//   B: 4x16  -> 2 VGPRs/lane: lane = N + 16*(K/2), vgpr = K%2
//   C/D: 16x16 -> 8 VGPRs/lane: vgpr j holds M = j + 8*(lane>=16), N = lane%16
// ---------------------------------------------------------------------------
__device__ __forceinline__ v8f wmma4(v2f a, v2f b, v8f c) {
  return __builtin_amdgcn_wmma_f32_16x16x4_f32(
      /*neg_a=*/false, a, /*neg_b=*/false, b,
      /*c_mod=*/(short)0, c, /*reuse_a=*/false, /*reuse_b=*/false);
}

__device__ __forceinline__ v2f frag_a(const float* base, int stride, int kk,
                                      int l, int hi) {
  v2f a;
  a.x = base[l * stride + kk + 2 * hi];
  a.y = base[l * stride + kk + 2 * hi + 1];
  return a;
}

__device__ __forceinline__ v2f frag_b(const float* Wrow0, int ldw, int kk,
                                      int l, int hi) {
  v2f b;
  b.x = Wrow0[(size_t)l * ldw + kk + 2 * hi];
  b.y = Wrow0[(size_t)l * ldw + kk + 2 * hi + 1];
  return b;
}

// Dual-tile K-loop: one A fragment feeds two WMMA (adjacent 16-col tiles).
// Prefetch (global_prefetch_b8) runs 512B ahead on both weight streams.
__device__ __forceinline__ void gemm_pair(const float* __restrict__ Alds,
                                          int astrd,
                                          const float* __restrict__ W0,
                                          const float* __restrict__ W1,
                                          int l, int hi, v8f& acc0, v8f& acc1) {
  for (int kk = 0; kk < D_; kk += 32) {
    __builtin_prefetch(W0 + (size_t)l * D_ + kk + 128, 0, 3);
    __builtin_prefetch(W1 + (size_t)l * D_ + kk + 128, 0, 3);
#pragma unroll
    for (int k4 = 0; k4 < 32; k4 += 4) {
      v2f a = frag_a(Alds, astrd, kk + k4, l, hi);
      acc0 = wmma4(a, frag_b(W0, D_, kk + k4, l, hi), acc0);
      acc1 = wmma4(a, frag_b(W1, D_, kk + k4, l, hi), acc1);
    }
  }
}

// ---------------------------------------------------------------------------
// Kernel A: embedding gather + K/V projection of the token stream.
//   X[bt, :]  = (ids[bt] != 0) ? embedding[ids[bt]] : 0
//   ky[bt, :] = X @ Wk^T + bk     (Wk = in_proj_w rows [512,1024))
//   vy[bt, :] = X @ Wv^T + bv     (Wv = in_proj_w rows [1024,1536))
// Grid: (8192/16, 4), block 256 (8 waves). Each wave: two 16x16 tiles, K=512.
// ---------------------------------------------------------------------------
__global__ __launch_bounds__(256)
void MVSEM_kv_embed(const int* __restrict__ ids, const float* __restrict__ emb,
                    const float* __restrict__ Win, const float* __restrict__ bin,
                    float* __restrict__ ky, float* __restrict__ vy) {
  __shared__ float Xs[16 * STRD_A];
  const int bt0 = blockIdx.x * 16;
  const int tid = threadIdx.x;

  for (int i = tid; i < 16 * D_; i += 256) {
    const int r = i >> 9, c = i & (D_ - 1);
    const int id = ids[bt0 + r];
    Xs[r * STRD_A + c] = (id != 0) ? emb[(size_t)id * D_ + c] : 0.0f;
  }
  __syncthreads();

  const int w = tid >> 5, lane = tid & 31, l = lane & 15, hi = lane >> 4;
  const int n = blockIdx.y * 256 + w * 32;        // 0..1023 over [K;V] columns
  const float* W0 = Win + (size_t)(D_ + n) * D_;
  const float* W1 = W0 + (size_t)16 * D_;

  v8f acc0, acc1;
  const float b0 = bin[D_ + n + l], b1 = bin[D_ + n + 16 + l];
#pragma unroll
  for (int j = 0; j < 8; ++j) { acc0[j] = b0; acc1[j] = b1; }

  gemm_pair(Xs, STRD_A, W0, W1, l, hi, acc0, acc1);

  float* dst = (n < D_) ? ky : vy;            // n is a multiple of 32, so both
  const int n0 = (n < D_) ? n : (n - D_);     // tiles land on the same side
#pragma unroll
  for (int j = 0; j < 8; ++j) {
    const int m = j + 8 * hi;
    dst[(size_t)(bt0 + m) * D_ + n0 + l]      = acc0[j];
    dst[(size_t)(bt0 + m) * D_ + n0 + 16 + l] = acc1[j];
  }
}

// ---------------------------------------------------------------------------
// Kernel B: the recurrent memory scan. One workgroup per *pair* of batches
// (M tile = 2 batches x 8 slots = 16 rows, exactly one WMMA row-tile).
// 8 blocks x 512 threads (16 waves), persistent over all 512 timesteps.
// LDS (dynamic): M | qkv (reused as resid) | o | token k/v rows | attn probs
// ---------------------------------------------------------------------------
__global__ __launch_bounds__(512)
void MVSEM_scan(const float* __restrict__ mem0,
                const float* __restrict__ Win,  const float* __restrict__ bin,
                const float* __restrict__ Wout, const float* __restrict__ bout,
                const float* __restrict__ lng,  const float* __restrict__ lnb,
                const float* __restrict__ ky,   const float* __restrict__ vy,
                float* __restrict__ MS) {
  extern __shared__ float sm[];
  float* Mld  = sm;                     // 16*STRD_A
  float* qkv  = Mld + 16 * STRD_A;      // 16*STRD_Q (cols 0..511 reused: resid)
  float* olds = qkv + 16 * STRD_Q;      // 16*STRD_A
  float* tok  = olds + 16 * STRD_A;     // 2 batches * (k row | v row) = 2048
  float* attn = tok + 2048;             // 16 waves * 72 probs

  const int tid = threadIdx.x;
  const int w = tid >> 5, lane = tid & 31, l = lane & 15, hi = lane >> 4;
  const int bpair = blockIdx.x;         // batches 2*bpair, 2*bpair+1

  for (int i = tid; i < 16 * D_; i += 512) {
    const int r = i >> 9, c = i & (D_ - 1);
    Mld[r * STRD_A + c] = mem0[(r & 7) * D_ + c];
  }
  __syncthreads();

  for (int t = 0; t < T_; ++t) {
    // ---- stage token k/v rows for both batches into LDS --------------------
    for (int i = tid; i < 2048; i += 512) {
      const int lb = i >> 10, r = (i >> 9) & 1, c = i & (D_ - 1);
      const size_t src = ((size_t)(2 * bpair + lb) * T_ + t) * D_ + c;
      tok[lb * 1024 + r * 512 + c] = r ? vy[src] : ky[src];
    }

    // ---- fused QKV GEMM: 16 x 1536, K = 512 (3 tile-pairs per wave) --------
    for (int ct = 2 * w; ct < 96; ct += 32) {
      const int n0 = ct * 16;
      const float* W0 = Win + (size_t)n0 * D_;
      const float* W1 = W0 + (size_t)16 * D_;
      v8f acc0, acc1;
      const float b0 = bin[n0 + l], b1 = bin[n0 + 16 + l];
#pragma unroll
      for (int j = 0; j < 8; ++j) { acc0[j] = b0; acc1[j] = b1; }
      gemm_pair(Mld, STRD_A, W0, W1, l, hi, acc0, acc1);
#pragma unroll
      for (int j = 0; j < 8; ++j) {
        const int m = j + 8 * hi;
        qkv[m * STRD_Q + n0 + l]      = acc0[j];
        qkv[m * STRD_Q + n0 + 16 + l] = acc1[j];
      }
    }
    __syncthreads();

    // ---- attention: wave w handles (local batch lb, head h) ----------------
    {
      const int lb = w >> 3, h = w & 7;
      const float* qbase = qkv + (lb * 8) * STRD_Q + h * DH_;
      const float* kbase = qbase + D_;        // K block at col offset 512
      const float* vbase = qbase + 2 * D_;    // V block at col offset 1024
      const float* kyrow = tok + lb * 1024 + h * DH_;
      const float* vyrow = tok + lb * 1024 + 512 + h * DH_;

      for (int i = 0; i < 8; ++i) {
        float s = -3.0e38f;
        if (lane < 9) {
          const float* krow = (lane < 8) ? (kbase + lane * STRD_Q) : kyrow;
          float d0 = 0.f;
          for (int d = 0; d < DH_; ++d) d0 += qbase[i * STRD_Q + d] * krow[d];
          s = d0 * 0.125f;                    // 1/sqrt(64)
        }
        float m = s;
#pragma unroll
        for (int off = 16; off > 0; off >>= 1) m = fmaxf(m, __shfl_xor(m, off, 32));
        const float e = (lane < 9) ? __expf(s - m) : 0.f;
        float su = e;
#pragma unroll
        for (int off = 16; off > 0; off >>= 1) su += __shfl_xor(su, off, 32);
        if (lane < 9) attn[w * 72 + i * 9 + lane] = e / su;
      }
      for (int i = 0; i < 8; ++i) {
#pragma unroll
        for (int part = 0; part < 2; ++part) {
          const int d = part * 32 + lane;
          float acc = attn[w * 72 + i * 9 + 8] * vyrow[d];
#pragma unroll
          for (int j = 0; j < 8; ++j)
            acc += attn[w * 72 + i * 9 + j] * vbase[j * STRD_Q + d];
          olds[(lb * 8 + i) * STRD_A + h * DH_ + d] = acc;
        }
      }
    }
    __syncthreads();

    // ---- output projection + residual: 16 x 512 (1 tile-pair per wave) -----
    {
      const int n0 = (2 * w) * 16;
      const float* W0 = Wout + (size_t)n0 * D_;
      const float* W1 = W0 + (size_t)16 * D_;
      v8f acc0, acc1;
      const float b0 = bout[n0 + l], b1 = bout[n0 + 16 + l];
#pragma unroll
      for (int j = 0; j < 8; ++j) { acc0[j] = b0; acc1[j] = b1; }
      gemm_pair(olds, STRD_A, W0, W1, l, hi, acc0, acc1);
#pragma unroll
      for (int j = 0; j < 8; ++j) {
        const int m = j + 8 * hi;
        qkv[m * STRD_Q + n0 + l]      = acc0[j] + Mld[m * STRD_A + n0 + l];
        qkv[m * STRD_Q + n0 + 16 + l] = acc1[j] + Mld[m * STRD_A + n0 + 16 + l];
      }
    }
    __syncthreads();

    // ---- LayerNorm: wave w normalizes row w --------------------------------
    {
      float vals[16];
      float su = 0.f, sq = 0.f;
#pragma unroll
      for (int ii = 0; ii < 16; ++ii) {
        const float x = qkv[w * STRD_Q + ii * 32 + lane];
        vals[ii] = x; su += x; sq += x * x;
      }
#pragma unroll
      for (int off = 16; off > 0; off >>= 1) {
        su += __shfl_xor(su, off, 32);
        sq += __shfl_xor(sq, off, 32);
      }
      const float mu  = su * (1.f / (float)D_);
      const float var = sq * (1.f / (float)D_) - mu * mu;
      const float inv = rsqrtf(var + LN_EPS);
#pragma unroll
      for (int ii = 0; ii < 16; ++ii) {
        const int c = ii * 32 + lane;
        Mld[w * STRD_A + c] = (vals[ii] - mu) * inv * lng[c] + lnb[c];
      }
    }
    __syncthreads();

    // ---- summaries: MS[b, t, :] = mean over 8 memory rows ------------------
    {
      const int c = tid;  // 512 threads == 512 columns
#pragma unroll
      for (int lb = 0; lb < 2; ++lb) {
        float s = 0.f;
#pragma unroll
        for (int r = 0; r < 8; ++r) s += Mld[(lb * 8 + r) * STRD_A + c];
        MS[(((size_t)(2 * bpair + lb)) * T_ + t) * D_ + c] = s * 0.125f;
      }
    }
    __syncthreads();
  }
}

// ---------------------------------------------------------------------------
// Kernel C: final projection, out = relu(MS @ proj_w^T + proj_b).
// (8192 x 512) @ (512 x 2048). Grid (512, 8), block 256 (8 waves);
// each wave two 16x16 output tiles, K = 512, A tile staged in LDS.
// ---------------------------------------------------------------------------
__global__ __launch_bounds__(256)
void MVSEM_proj(const float* __restrict__ MS, const float* __restrict__ Wp,
                const float* __restrict__ bp, float* __restrict__ out) {
  __shared__ float Xs[16 * STRD_A];
  const int bt0 = blockIdx.x * 16;
  const int tid = threadIdx.x;
  for (int i = tid; i < 16 * D_; i += 256) {
    const int r = i >> 9, c = i & (D_ - 1);
    Xs[r * STRD_A + c] = MS[(size_t)(bt0 + r) * D_ + c];
  }
  __syncthreads();

  const int w = tid >> 5, lane = tid & 31, l = lane & 15, hi = lane >> 4;
  const int n0 = blockIdx.y * 256 + w * 32;
  const float* W0 = Wp + (size_t)n0 * D_;
  const float* W1 = W0 + (size_t)16 * D_;

  v8f acc0, acc1;
  const float b0 = bp[n0 + l], b1 = bp[n0 + 16 + l];
#pragma unroll
  for (int j = 0; j < 8; ++j) { acc0[j] = b0; acc1[j] = b1; }

  gemm_pair(Xs, STRD_A, W0, W1, l, hi, acc0, acc1);

#pragma unroll
  for (int j = 0; j < 8; ++j) {
    const int m = j + 8 * hi;
    out[(size_t)(bt0 + m) * OUT_ + n0 + l]      = fmaxf(acc0[j], 0.0f);
    out[(size_t)(bt0 + m) * OUT_ + n0 + 16 + l] = fmaxf(acc1[j], 0.0f);
  }
}

// ---------------------------------------------------------------------------
extern "C" void kernel_launch(void* const* d_in, const int* in_sizes, int n_in,
                              void* d_out, int out_size, void* d_ws, size_t ws_size,
                              hipStream_t stream) {
  const int*   ids  = (const int*)d_in[0];     // (B,T)
  const float* emb  = (const float*)d_in[1];   // (V,D)
  const float* mem0 = (const float*)d_in[2];   // (NM,D)
  const float* Win  = (const float*)d_in[3];   // (3D,D)
  const float* bin  = (const float*)d_in[4];   // (3D,)
  const float* Wout = (const float*)d_in[5];   // (D,D)
  const float* bout = (const float*)d_in[6];   // (D,)
  const float* lng  = (const float*)d_in[7];   // (D,)
  const float* lnb  = (const float*)d_in[8];   // (D,)
  const float* Wp   = (const float*)d_in[9];   // (OUT,D)
  const float* bp   = (const float*)d_in[10];  // (OUT,)
  float* out = (float*)d_out;                  // (B,T,OUT)

  const size_t btd = (size_t)B_ * T_ * D_;
  float* ky = (float*)d_ws;        // (B,T,D) token K projections
  float* vy = ky + btd;            // (B,T,D) token V projections
  float* MS = vy + btd;            // (B,T,D) per-step memory summaries

  // A: token K/V projections (parallel, hoisted out of the scan)
  dim3 gA(B_ * T_ / 16, 4);
  MVSEM_kv_embed<<<gA, 256, 0, stream>>>(ids, emb, Win, bin, ky, vy);

  // B: the recurrent scan, 8 persistent workgroups (2 batches each)
  const size_t smB =
      (size_t)(16 * STRD_A + 16 * STRD_Q + 16 * STRD_A + 2048 + 16 * 72) *
      sizeof(float);
  hipFuncSetAttribute(reinterpret_cast<const void*>(MVSEM_scan),
                      hipFuncAttributeMaxDynamicSharedMemorySize, (int)smB);
  MVSEM_scan<<<B_ / 2, 512, smB, stream>>>(mem0, Win, bin, Wout, bout, lng, lnb,
                                           ky, vy, MS);

  // C: final projection + ReLU
  dim3 gC(B_ * T_ / 16, OUT_ / 256);
  MVSEM_proj<<<gC, 256, 0, stream>>>(MS, Wp, bp, out);
}